// CTRPredictor_7687991460132
// MI455X (gfx1250) — compile-verified
//
#include <hip/hip_runtime.h>
#include <hip/hip_bf16.h>

typedef float v2f __attribute__((ext_vector_type(2)));
typedef float v8f __attribute__((ext_vector_type(8)));

#define D 128
#define EPS 1e-12f

// ---------------------------------------------------------------------------
// Kernel 1: per-row inverse L2 norm. One wave32 per row; each lane owns 4
// consecutive floats (float4 load), wave-wide butterfly reduction.
// ---------------------------------------------------------------------------
__global__ void row_rinv_kernel(const float* __restrict__ x,
                                float* __restrict__ rinv, int N) {
    const int lane = threadIdx.x & 31;
    const int row  = blockIdx.x * (blockDim.x >> 5) + (threadIdx.x >> 5);
    if (row >= N) return;  // uniform per wave

    const float4* p = reinterpret_cast<const float4*>(x + (size_t)row * D);
    float4 v = p[lane];
    float ss = v.x * v.x + v.y * v.y + v.z * v.z + v.w * v.w;
#pragma unroll
    for (int off = 16; off > 0; off >>= 1)
        ss += __shfl_xor(ss, off, 32);
    if (lane == 0)
        rinv[row] = 1.0f / fmaxf(sqrtf(ss), EPS);
}

// ---------------------------------------------------------------------------
// Kernel 2: 16 edges per wave via V_WMMA_F32_16X16X4_F32.
//   A (16x4 f32): row m = 4-wide chunk of x[src[base+m]]
//     lanes 0-15 hold K=0 (VGPR0), K=1 (VGPR1); lanes 16-31 hold K=2,3.
//   B (4x16 f32): col n = 4-wide chunk of x[dst[base+n]]
//     same per-lane addressing (lane l -> row (l&15), cols k + 2*(l>>4)).
//   After 32 K-steps the 16 pair-dots sit on the diagonal of the 16x16 C:
//     e in 0..7  -> lane e,    VGPR e
//     e in 8..15 -> lane e+16, VGPR e-8   (i.e. lanes 24..31, VGPR lane-24)
// EXEC stays all-ones through every WMMA (edge index clamped); predication
// only guards the final store.
// ---------------------------------------------------------------------------
__global__ void edge_score_kernel(const float* __restrict__ x,
                                  const float* __restrict__ rinv,
                                  const int* __restrict__ src,
                                  const int* __restrict__ dst,
                                  float* __restrict__ out, int E) {
    const int lane = threadIdx.x & 31;
    const int wave = (blockIdx.x * blockDim.x + threadIdx.x) >> 5;
    const int base = wave * 16;
    if (base >= E) return;  // uniform per wave

    const int eoff = lane & 15;
    const int e    = min(base + eoff, E - 1);  // clamp keeps EXEC full
    const int s    = src[e];
    const int d    = dst[e];

    const float* __restrict__ prow_s = x + (size_t)s * D;
    const float* __restrict__ prow_d = x + (size_t)d * D;
    const int colsel = (lane >> 4) << 1;  // 0 for lanes 0-15, 2 for 16-31

    v8f acc = {0.f, 0.f, 0.f, 0.f, 0.f, 0.f, 0.f, 0.f};
#pragma unroll
    for (int k = 0; k < D; k += 4) {
        v2f a = *reinterpret_cast<const v2f*>(prow_s + k + colsel);
        v2f b = *reinterpret_cast<const v2f*>(prow_d + k + colsel);
        // 8 args: (neg_a, A, neg_b, B, c_mod, C, reuse_a, reuse_b)
        acc = __builtin_amdgcn_wmma_f32_16x16x4_f32(
            false, a, false, b, (short)0, acc, false, false);
    }

    // Diagonal extraction: lanes 0-7 and 24-31 each own one edge's score.
    const bool active = (lane < 8) || (lane >= 24);
    const int  idx    = (lane < 8) ? lane : (lane - 24);
    float val = acc[0];
#pragma unroll
    for (int j = 1; j < 8; ++j)
        if (idx == j) val = acc[j];

    if (active) {
        const int eo = base + eoff;
        if (eo < E) out[eo] = val * rinv[s] * rinv[d];
    }
}

// ---------------------------------------------------------------------------
// Launch: inputs are (x[N*128] f32, src[E] i32, dst[E] i32); output E f32.
// d_ws holds N floats of per-row inverse norms (400 KB << ws_size).
// ---------------------------------------------------------------------------
extern "C" void kernel_launch(void* const* d_in, const int* in_sizes, int n_in,
                              void* d_out, int out_size, void* d_ws, size_t ws_size,
                              hipStream_t stream) {
    const float* x   = (const float*)d_in[0];
    const int*   src = (const int*)d_in[1];
    const int*   dst = (const int*)d_in[2];
    float*       out = (float*)d_out;
    float*       rinv = (float*)d_ws;

    const int N = in_sizes[0] / D;
    const int E = in_sizes[1];

    // Kernel 1: 8 rows per 256-thread block (8 waves).
    const int rows_per_block = 256 / 32;
    row_rinv_kernel<<<(N + rows_per_block - 1) / rows_per_block, 256, 0, stream>>>(
        x, rinv, N);

    // Kernel 2: 8 waves per block * 16 edges per wave = 128 edges per block.
    const int edges_per_block = (256 / 32) * 16;
    edge_score_kernel<<<(E + edges_per_block - 1) / edges_per_block, 256, 0, stream>>>(
        x, rinv, src, dst, out, E);
}